// Model_74302934221566
// MI455X (gfx1250) — compile-verified
//
#include <hip/hip_runtime.h>
#include <hip/hip_bf16.h>

typedef __attribute__((ext_vector_type(2))) float v2f;
typedef __attribute__((ext_vector_type(8))) float v8f;

#define NPIX 65536   // 256*256 pixels
#define NB   6       // batch

// Order-preserving float<->uint encoding: unsigned compare order == float order.
__device__ __forceinline__ unsigned ord_enc(float f) {
  unsigned u = __float_as_uint(f);
  return (u & 0x80000000u) ? ~u : (u | 0x80000000u);
}
__device__ __forceinline__ float ord_dec(unsigned e) {
  unsigned u = (e & 0x80000000u) ? (e ^ 0x80000000u) : ~e;
  return __uint_as_float(u);
}

__global__ void init_red(unsigned* red) {
  red[0] = 0xFF800000u;  // enc(+inf): running-MIN slot
  red[1] = 0x007FFFFFu;  // enc(-inf): running-MAX slot
}

// Kernel 1: per-pixel 3x3 channel mix via V_WMMA_F32_16X16X4_F32,
// batch-max into M[c][h][w], global min/max of v2 via wave reduce + atomics.
// Wave handles 16 pixels: A = W^T (16x4, rows 0-2 live, K=3 column zero),
// B = x tile (4x16), C = bias+3 broadcast.
// D row m, lanes 0-15 = v2[channel m] for the 16 pixels.
__global__ __launch_bounds__(256) void channel_mix_max(
    const float* __restrict__ x, const float* __restrict__ Wp,
    const float* __restrict__ bp, float* __restrict__ M,
    unsigned* __restrict__ red)
{
  const int lane = threadIdx.x & 31;
  const int wave = threadIdx.x >> 5;
  const int pixBase = (blockIdx.x * 8 + wave) << 4;
  const int m = lane & 15;
  const bool lo16 = lane < 32 ? (lane < 16) : false;

  // A (16x4): A[m][k] = W[k][m] (einsum 'bihw,io'); ISA layout:
  // lanes 0-15: v0=K0, v1=K1 ; lanes 16-31: v0=K2, v1=K3(pad=0)
  v2f a;
  a.x = (m < 3) ? (lo16 ? Wp[m] : Wp[6 + m]) : 0.f;
  a.y = (m < 3 && lo16) ? Wp[3 + m] : 0.f;

  // C (16x16): rows 0..2 = bias[c] + 3 (folds v1 bias and v2's +3)
  v8f c = {};
  c[0] = lo16 ? (bp[0] + 3.f) : 0.f;
  c[1] = lo16 ? (bp[1] + 3.f) : 0.f;
  c[2] = lo16 ? (bp[2] + 3.f) : 0.f;

  const int pix = pixBase + m;
  // B layout: lanes 0-15: v0=K0 row, v1=K1 row; lanes 16-31: v0=K2 row, v1=K3 row.
  // K=3 row of B is a don't-care (A column 3 is zero), so lanes>=16 load the
  // channel-1 value unconditionally -- no divergent branch around the load.
  const int ch0 = lo16 ? 0 : 2;
  float vm0 = -INFINITY, vm1 = -INFINITY, vm2 = -INFINITY;
  float lmin = INFINITY;

#pragma unroll
  for (int b = 0; b < NB; ++b) {
    v2f bm;
    bm.x = x[(size_t)(b * 3 + ch0) * NPIX + pix];
    bm.y = x[(size_t)(b * 3 + 1) * NPIX + pix];   // K1 (lanes<16) / K3 don't-care
    v8f d = __builtin_amdgcn_wmma_f32_16x16x4_f32(
        /*neg_a=*/false, a, /*neg_b=*/false, bm,
        /*c_mod=*/(short)0, c, /*reuse_a=*/false, /*reuse_b=*/false);
    vm0 = fmaxf(vm0, d[0]);
    vm1 = fmaxf(vm1, d[1]);
    vm2 = fmaxf(vm2, d[2]);
    lmin = fminf(lmin, fminf(d[0], fminf(d[1], d[2])));
  }

  // Wave32 reduction of global min/max of v2 (lanes >=16 hold zero rows -> exclude)
  float lo = lo16 ? lmin : INFINITY;
  float hi = lo16 ? fmaxf(vm0, fmaxf(vm1, vm2)) : -INFINITY;
  for (int off = 16; off > 0; off >>= 1) {
    lo = fminf(lo, __shfl_xor(lo, off, 32));
    hi = fmaxf(hi, __shfl_xor(hi, off, 32));
  }
  if (lane == 0) {
    atomicMin(&red[0], ord_enc(lo));
    atomicMax(&red[1], ord_enc(hi));
  }

  if (lo16) {
    M[pix]             = vm0;
    M[NPIX + pix]      = vm1;
    M[2 * NPIX + pix]  = vm2;
  }
}

// Kernel 2: out[p,q,r] = M[q%3, p%256, r%256] + S.  One block per (p,q) row
// of 768; 192 threads x float4. Pure streaming write (21.2 MB) — the roofline
// limiter; M rows (768 KB total) stay L2-resident.
__global__ __launch_bounds__(192) void broadcast_out(
    const float* __restrict__ M, const unsigned* __restrict__ red,
    float* __restrict__ out)
{
  const int row = blockIdx.x;        // row = p*9 + q, 6912 rows
  const int p = row / 9;
  const int q = row - p * 9;
  const int ch = q % 3;

  // Scalar S from global min/max (all other reductions are monotone fns of v2)
  const float gmin = ord_dec(red[0]);
  const float gmax = ord_dec(red[1]);
  const float f6min = fmaxf(3.f * gmin, 0.f) + 3.f;   // min(v6) == min(v47) == min(v33)
  const float f6max = fmaxf(3.f * gmax, 0.f) + 3.f;
  const float maxv8 = 3.f * f6max + 3.f;              // max(v8)
  float t = 3.f * f6min - 3.f;                        // min(v21) via relu chain
  t = fmaxf(t, 0.f) - 3.f;
  t = fmaxf(t, 0.f) - 3.f;
  t = fmaxf(t, 0.f) - 3.f;
  t -= 3.f;
  const float S = gmax + maxv8 + t + 2.f * f6min;

  const float* mrow = M + ((size_t)ch << 16) + ((size_t)(p & 255) << 8);
  const int col = threadIdx.x << 2;                   // 0..764
  float4 v = *(const float4*)(mrow + (col & 255));    // r%256 (4 | 256, no wrap)
  v.x += S; v.y += S; v.z += S; v.w += S;
  *(float4*)(out + (size_t)row * 768 + col) = v;
}

extern "C" void kernel_launch(void* const* d_in, const int* in_sizes, int n_in,
                              void* d_out, int out_size, void* d_ws, size_t ws_size,
                              hipStream_t stream) {
  const float* x  = (const float*)d_in[0];   // (6,3,256,256) f32
  const float* W  = (const float*)d_in[1];   // (3,3) f32, [in][out]
  const float* b  = (const float*)d_in[2];   // (3,) f32
  float* out = (float*)d_out;                // 768*9*768 f32

  unsigned* red = (unsigned*)d_ws;                    // 2 x u32 min/max slots
  float* M = (float*)((char*)d_ws + 256);             // 3*65536 f32 = 768 KB

  init_red<<<1, 1, 0, stream>>>(red);
  channel_mix_max<<<512, 256, 0, stream>>>(x, W, b, M, red);  // 512*8 waves * 16 pix = 65536
  broadcast_out<<<6912, 192, 0, stream>>>(M, red, out);       // 768*9 rows
}